// ROIPool3d_6725918786268
// MI455X (gfx1250) — compile-verified
//
#include <hip/hip_runtime.h>
#include <hip/hip_bf16.h>

#define PH_ 7
#define PW_ 7
#define SCALE_ 0.0625f
#define B_ 2
#define C_ 64
#define L_ 8
#define H_ 64
#define W_ 64
#define NROIS_ 32

// outputs per roi = C*L*PH*PW = 64*8*49 = 25088 = 98 * 256
// -> each 256-thread block lies entirely within one ROI.

__global__ __launch_bounds__(256) void ROIPool3d_kernel(
    const float* __restrict__ input,   // (B, C, L, H, W)
    const float* __restrict__ rois,    // (NROIS, 5): b, x1, y1, x2, y2
    float* __restrict__ out,           // (NROIS, C, L, PH, PW)
    int total)
{
    const int tid  = threadIdx.x;
    const int gidx = blockIdx.x * 256 + tid;

    const int per_roi = C_ * L_ * PH_ * PW_;       // 25088
    const int n = gidx / per_roi;                  // same for whole block

    // ---- Stage this block's ROI descriptor (5 floats) into LDS via the
    //      CDNA5 async global->LDS path (tracked by ASYNCcnt). ----
    __shared__ float s_roi[8];
    if (tid < 5) {
        unsigned lds_addr = (unsigned)(size_t)(&s_roi[tid]);          // low 32b = LDS byte offset
        const float* gptr = rois + n * 5 + tid;
        asm volatile("global_load_async_to_lds_b32 %0, %1, off"
                     :: "v"(lds_addr), "v"(gptr) : "memory");
    }
    asm volatile("s_wait_asynccnt 0" ::: "memory");
    __syncthreads();

    if (gidx >= total) return;

    const int rem  = gidx % per_roi;
    const int c    = rem / (L_ * PH_ * PW_);
    const int rem2 = rem % (L_ * PH_ * PW_);
    const int l    = rem2 / (PH_ * PW_);
    const int rem3 = rem2 % (PH_ * PW_);
    const int ph   = rem3 / PW_;
    const int pw   = rem3 % PW_;

    // ---- Bin geometry (f32 math matching the JAX reference; rintf = RNE
    //      like jnp.round; true f32 division by 7). ----
    const int b  = (int)s_roi[0];
    const int sw = (int)rintf(s_roi[1] * SCALE_);
    const int sh = (int)rintf(s_roi[2] * SCALE_);
    const int ew = (int)rintf(s_roi[3] * SCALE_);
    const int eh = (int)rintf(s_roi[4] * SCALE_);

    const float roi_w = (float)max(ew - sw + 1, 1);
    const float roi_h = (float)max(eh - sh + 1, 1);
    const float bin_h = roi_h / 7.0f;
    const float bin_w = roi_w / 7.0f;

    int hstart = (int)floorf((float)ph * bin_h) + sh;
    int hend   = (int)ceilf((float)(ph + 1) * bin_h) + sh;
    hstart = min(max(hstart, 0), H_);
    hend   = min(max(hend,   0), H_);

    int wstart = (int)floorf((float)pw * bin_w) + sw;
    int wend   = (int)ceilf((float)(pw + 1) * bin_w) + sw;
    wstart = min(max(wstart, 0), W_);
    wend   = min(max(wend,   0), W_);

    float result = 0.0f;
    if (hstart < hend && wstart < wend) {
        const float* base =
            input + (((size_t)b * C_ + c) * L_ + l) * (size_t)(H_ * W_);
        float m = -__FLT_MAX__;
        const int nw = wend - wstart;
        for (int h = hstart; h < hend; ++h) {
            const float* row = base + h * W_ + wstart;
            if (h + 1 < hend) {
                __builtin_prefetch(row + W_, 0, 0);   // global_prefetch_b8 on next row
            }
            #pragma unroll 4
            for (int w = 0; w < nw; ++w) {
                m = fmaxf(m, row[w]);
            }
        }
        result = m;
    }

    out[gidx] = result;
}

extern "C" void kernel_launch(void* const* d_in, const int* in_sizes, int n_in,
                              void* d_out, int out_size, void* d_ws, size_t ws_size,
                              hipStream_t stream) {
    const float* input = (const float*)d_in[0];   // (2,64,8,64,64) f32
    const float* rois  = (const float*)d_in[1];   // (32,5) f32
    float* out         = (float*)d_out;           // (32,64,8,7,7) f32

    const int total  = out_size;                  // 802816
    const int blocks = (total + 255) / 256;       // 3136 (exact)
    ROIPool3d_kernel<<<blocks, 256, 0, stream>>>(input, rois, out, total);
}